// SelfAttn_41781441856090
// MI455X (gfx1250) — compile-verified
//
#include <hip/hip_runtime.h>
#include <hip/hip_bf16.h>
#include <math.h>

#define H 256
#define L 256
#define B 8
#define M_ROWS (B * L) /* 2048 rows of the (B*L, H) projection */

typedef float v2f __attribute__((ext_vector_type(2)));
typedef float v8f __attribute__((ext_vector_type(8)));

__device__ __forceinline__ float fast_tanh(float x) {
#if __has_builtin(__builtin_amdgcn_tanhf)
    return __builtin_amdgcn_tanhf(x);
#else
    float y;
    asm volatile("v_tanh_f32 %0, %1\n\ts_nop 1" : "=v"(y) : "v"(x));
    return y;
#endif
}

// ---------------------------------------------------------------------------
// Kernel A: P1[b*L+i, h] = (h_row . W1[h,:]) + b_attn[h]
//           P2[b*L+j, h] = (h_row . W2[h,:])
// One wave (32 lanes) per 16x16 output tile; V_WMMA_F32_16X16X4_F32, K=4/step.
// A-frag (16x4 f32): lane m = lane&15, k-pair = (lane>>4)*2  -> 2 VGPRs
// B-frag (4x16 f32): lane n = lane&15, k-pair = (lane>>4)*2  -> 2 VGPRs
// C/D  (16x16 f32): col = lane&15, rows = (lane>>4)*8 + r    -> 8 VGPRs
// ---------------------------------------------------------------------------
__global__ void __launch_bounds__(32) proj_wmma_kernel(
    const float* __restrict__ aph,    // (L, B, H)
    const float* __restrict__ Wattn,  // (H, 2H) row-major
    const float* __restrict__ battn,  // (H)
    float* __restrict__ P1,           // (B*L, H)
    float* __restrict__ P2)           // (B*L, H)
{
    const int lane  = threadIdx.x;      // 0..31, full wave, EXEC all ones
    const int mtile = blockIdx.x >> 4;  // 0..127
    const int ntile = blockIdx.x & 15;  // 0..15

    const int l16 = lane & 15;
    const int kb  = (lane >> 4) << 1;   // 0 or 2

    // A: row m of h (B,L,H);  h[b,l,:] == aph[l,b,:]
    const int m  = mtile * 16 + l16;
    const int bb = m >> 8;              // m / L
    const int ll = m & 255;             // m % L
    const float* hrow = aph + (size_t)(ll * B + bb) * H;

    // B: output column ncol selects a row of W_attn (W1|W2 concatenated)
    const int ncol = ntile * 16 + l16;
    const float* wrow = Wattn + (size_t)ncol * (2 * H);

    v8f acc1 = {};
    v8f acc2 = {};
#pragma unroll 4
    for (int k0 = 0; k0 < H; k0 += 4) {
        v2f a  = *(const v2f*)(hrow + k0 + kb);
        v2f b1 = *(const v2f*)(wrow + k0 + kb);
        v2f b2 = *(const v2f*)(wrow + H + k0 + kb);
        acc1 = __builtin_amdgcn_wmma_f32_16x16x4_f32(
            /*neg_a=*/false, a, /*neg_b=*/false, b1,
            /*c_mod=*/(short)0, acc1, /*reuse_a=*/false, /*reuse_b=*/false);
        acc2 = __builtin_amdgcn_wmma_f32_16x16x4_f32(
            false, a, false, b2, (short)0, acc2, false, false);
    }

    const float bias = battn[ncol];
    const int row0 = mtile * 16 + ((lane >> 4) << 3);
#pragma unroll
    for (int r = 0; r < 8; ++r) {
        const size_t off = (size_t)(row0 + r) * H + ncol;
        P1[off] = acc1[r] + bias;   // fold b_attn into P1
        P2[off] = acc2[r];
    }
}

// ---------------------------------------------------------------------------
// Kernel B: one block per (b, j).  Thread i computes
//   e[i] = sum_h tanh(P1[b,i,h] + P2[b,j,h]) * W_score[h] + b_score
// then block-wide softmax over i, writing out[b, i, j].
// Dominant cost: 256 native v_tanh_f32 + FMA per thread (134M total).
// ---------------------------------------------------------------------------
__global__ void __launch_bounds__(256) energy_softmax_kernel(
    const float* __restrict__ P1,
    const float* __restrict__ P2,
    const float* __restrict__ Wscore,  // (H)
    const float* __restrict__ bscore,  // (1)
    float* __restrict__ out)           // (B, L, L)
{
    __shared__ __align__(16) float p2s[H];
    __shared__ __align__(16) float wss[H];
    __shared__ float sred[256];

    const int tid = threadIdx.x;
    const int bb  = blockIdx.x >> 8;   // b
    const int j   = blockIdx.x & 255;  // j

    p2s[tid] = P2[(size_t)(bb * L + j) * H + tid];
    wss[tid] = Wscore[tid];
    __syncthreads();

    const int i = tid;
    const float4* p1v = (const float4*)(P1 + (size_t)(bb * L + i) * H);
    const float4* p2v = (const float4*)p2s;
    const float4* wsv = (const float4*)wss;

    float acc = 0.0f;
#pragma unroll 4
    for (int hh = 0; hh < H / 4; ++hh) {
        const float4 a = p1v[hh];
        const float4 c = p2v[hh];
        const float4 w = wsv[hh];
        acc = fmaf(fast_tanh(a.x + c.x), w.x, acc);
        acc = fmaf(fast_tanh(a.y + c.y), w.y, acc);
        acc = fmaf(fast_tanh(a.z + c.z), w.z, acc);
        acc = fmaf(fast_tanh(a.w + c.w), w.w, acc);
    }
    const float e = acc + bscore[0];

    // ---- softmax over i (the 256 threads of this block) ----
    sred[tid] = e;
    __syncthreads();
    for (int s = 128; s > 0; s >>= 1) {
        if (tid < s) sred[tid] = fmaxf(sred[tid], sred[tid + s]);
        __syncthreads();
    }
    const float mx = sred[0];
    __syncthreads();

    const float ex = __expf(e - mx);
    sred[tid] = ex;
    __syncthreads();
    for (int s = 128; s > 0; s >>= 1) {
        if (tid < s) sred[tid] += sred[tid + s];
        __syncthreads();
    }
    const float sum = sred[0];

    out[(size_t)bb * L * L + (size_t)i * L + j] = ex / sum;
}

extern "C" void kernel_launch(void* const* d_in, const int* in_sizes, int n_in,
                              void* d_out, int out_size, void* d_ws, size_t ws_size,
                              hipStream_t stream) {
    const float* aph    = (const float*)d_in[0]; // (L,B,H)
    /* d_in[1] problem_num_mask: unused by the reference */
    const float* Wattn  = (const float*)d_in[2]; // (H,2H)
    const float* battn  = (const float*)d_in[3]; // (H)
    const float* Wscore = (const float*)d_in[4]; // (1,H)
    const float* bscore = (const float*)d_in[5]; // (1)
    float* out = (float*)d_out;                  // (B,L,L)

    float* P1 = (float*)d_ws;                    // 2048*256 f32 = 2 MB
    float* P2 = P1 + (size_t)M_ROWS * H;         // 2 MB

    // 128 M-tiles x 16 N-tiles, one wave each
    proj_wmma_kernel<<<dim3((M_ROWS / 16) * (H / 16)), dim3(32), 0, stream>>>(
        aph, Wattn, battn, P1, P2);

    // one block per (b, j)
    energy_softmax_kernel<<<dim3(B * L), dim3(256), 0, stream>>>(
        P1, P2, Wscore, bscore, out);
}